// LFQQuantizer_867583393934
// MI455X (gfx1250) — compile-verified
//
#include <hip/hip_runtime.h>
#include <hip/hip_bf16.h>

typedef float v2f __attribute__((ext_vector_type(2)));
typedef float v8f __attribute__((ext_vector_type(8)));

#define ROWS 4096            // 2 * 2048 rows
#define DIM 14               // codebook dim
#define KSIZE 16384          // codebook size
#define INV_T_2 200.0f       // 2 / TEMPERATURE
#define EPSV 1e-5f

// d_out layout (floats, concatenated in reference return order)
#define OFF_QUANT  0         // 57344
#define OFF_IDX    57344     // 4096
#define OFF_PSE    61440     // per_sample_entropy
#define OFF_CBE    61441     // codebook_entropy
#define OFF_AUX    61442     // entropy_aux_loss
#define OFF_COMMIT 61443     // commit_loss

__global__ void lfq_zero(float* out) {
    if (threadIdx.x < 4) out[OFF_PSE + threadIdx.x] = 0.0f;
}

// Per-row pass: sign quantize, bit-pack indices (big-endian), commit loss,
// and the exact factorized log-partition logZ[row] = sum_d (|a|+log1p(e^-2|a|)).
__global__ void __launch_bounds__(256)
lfq_rowwise(const float* __restrict__ x, float* __restrict__ out,
            float* __restrict__ logZ) {
    __shared__ float red[256];
    int row = blockIdx.x * blockDim.x + threadIdx.x;
    float cl = 0.0f;
    if (row < ROWS) {
        const float* xr = x + row * DIM;
        int idx = 0;
        float lz = 0.0f;
#pragma unroll
        for (int d = 0; d < DIM; ++d) {
            float xv = xr[d];
            bool pos = xv > 0.0f;
            float q = pos ? 1.0f : -1.0f;
            out[OFF_QUANT + row * DIM + d] = q;
            idx |= (pos ? 1 : 0) << (DIM - 1 - d);   // big-endian pack
            float df = xv - q;
            cl += df * df;
            float a = INV_T_2 * xv;
            float t = fabsf(a);
            lz += t + log1pf(__expf(-2.0f * t));
        }
        out[OFF_IDX + row] = (float)idx;
        logZ[row] = lz;
    }
    red[threadIdx.x] = cl;
    __syncthreads();
    for (int s = 128; s > 0; s >>= 1) {
        if ((int)threadIdx.x < s) red[threadIdx.x] += red[threadIdx.x + s];
        __syncthreads();
    }
    if (threadIdx.x == 0)
        atomicAdd(&out[OFF_COMMIT], red[0] * (1.0f / (float)(ROWS * DIM)));
}

// Heavy pass: scaled logits via chained V_WMMA_F32_16X16X4_F32 (K=14 padded
// to 16 with zero A columns; B synthesized from code bit patterns, exact ±1).
// probs = exp(scaled - logZ) single-pass (logZ exact from factorization).
// Each wave owns one 16-wide column tile -> writes avg_probs without atomics.
__global__ void __launch_bounds__(256)
lfq_entropy_wmma(const float* __restrict__ x, const float* __restrict__ logZ,
                 float* __restrict__ avg_probs, float* __restrict__ out) {
    const int lane = threadIdx.x & 31;
    const int wave = threadIdx.x >> 5;
    const int ct   = blockIdx.x * 8 + wave;   // column tile 0..1023
    const int m    = lane & 15;
    const int hi   = lane >> 4;
    const int jcol = ct * 16 + m;

    // B tile: 4x16 per K-step; VGPR v holds rows K = 4s + v + 2*hi, col = m.
    v2f b[4];
#pragma unroll
    for (int s = 0; s < 4; ++s) {
        int k0 = 4 * s + 2 * hi;
        b[s].x = ((jcol >> k0) & 1) ? 1.0f : -1.0f;
        b[s].y = ((jcol >> (k0 + 1)) & 1) ? 1.0f : -1.0f;
    }

    float colsum = 0.0f;
    float entAcc = 0.0f;

    for (int rt = 0; rt < ROWS / 16; ++rt) {
        // A tile (16x4 per step): lane row M=m, VGPR pair covers K=k0,k0+1,
        // scaled by 2/T; K>=14 padded with zeros.
        const float* xr = x + (rt * 16 + m) * DIM;
        v2f a[4];
#pragma unroll
        for (int s = 0; s < 4; ++s) {
            int k0 = 4 * s + 2 * hi;
            a[s].x = (k0 < DIM)     ? INV_T_2 * xr[k0]     : 0.0f;
            a[s].y = (k0 + 1 < DIM) ? INV_T_2 * xr[k0 + 1] : 0.0f;
        }

        v8f c = {};
#pragma unroll
        for (int s = 0; s < 4; ++s)
            c = __builtin_amdgcn_wmma_f32_16x16x4_f32(
                    false, a[s], false, b[s], (short)0, c, false, false);

        // C/D layout: VGPR v = row (v + 8*hi), col = m.
#pragma unroll
        for (int v = 0; v < 8; ++v) {
            float lz = logZ[rt * 16 + v + 8 * hi];
            float sc = c[v];
            float p  = __expf(sc - lz);       // exact softmax prob, <= 1
            colsum  += p;
            entAcc  += p * (lz - sc);         // -p * log(p)
        }
    }

    // columns: fold the two half-wave row groups together
    colsum += __shfl_xor(colsum, 16, 32);
    if (lane < 16) avg_probs[jcol] = colsum * (1.0f / (float)ROWS);

    // per-sample entropy: full wave reduce, one atomic per wave
#pragma unroll
    for (int off = 16; off > 0; off >>= 1)
        entAcc += __shfl_xor(entAcc, off, 32);
    if (lane == 0) atomicAdd(&out[OFF_PSE], entAcc * (1.0f / (float)ROWS));
}

__global__ void __launch_bounds__(256)
lfq_finalize(const float* __restrict__ avg_probs, float* out) {
    __shared__ float red[256];
    float acc = 0.0f;
    for (int i = threadIdx.x; i < KSIZE; i += 256) {
        float a = avg_probs[i];
        acc += a * __logf(a + EPSV);
    }
    red[threadIdx.x] = acc;
    __syncthreads();
    for (int s = 128; s > 0; s >>= 1) {
        if ((int)threadIdx.x < s) red[threadIdx.x] += red[threadIdx.x + s];
        __syncthreads();
    }
    if (threadIdx.x == 0) {
        float ce = -red[0];
        out[OFF_CBE] = ce;
        out[OFF_AUX] = out[OFF_PSE] - ce;   // 1.0*pse - 1.0*ce
    }
}

extern "C" void kernel_launch(void* const* d_in, const int* in_sizes, int n_in,
                              void* d_out, int out_size, void* d_ws, size_t ws_size,
                              hipStream_t stream) {
    const float* x = (const float*)d_in[0];
    // d_in[1] (codebook) unused: rows are exactly the +-1 bit patterns of the
    // code index, synthesized in-register inside lfq_entropy_wmma.
    float* out       = (float*)d_out;
    float* wsf       = (float*)d_ws;
    float* avg_probs = wsf;            // 16384 floats
    float* logZ      = wsf + KSIZE;    // 4096 floats

    lfq_zero<<<1, 32, 0, stream>>>(out);
    lfq_rowwise<<<ROWS / 256, 256, 0, stream>>>(x, out, logZ);
    lfq_entropy_wmma<<<128, 256, 0, stream>>>(x, logZ, avg_probs, out);
    lfq_finalize<<<1, 256, 0, stream>>>(avg_probs, out);
}